// DiffusionPropagate_35424890258068
// MI455X (gfx1250) — compile-verified
//
#include <hip/hip_runtime.h>

typedef __attribute__((ext_vector_type(2))) float v2f;
typedef __attribute__((ext_vector_type(8))) float v8f;

#define NN      4096
#define BB      8
#define NSEEDS  80
#define WAVES   16
#define KWAVE   (NN / WAVES)   // 256 K-values per wave (deep split-K)
#define PF_DIST 32             // prefetch 8 K-steps (32 P-rows) ahead

// ---- init: copy preds into workspace ping buffer, zero seed mask ----------
__global__ void diff_init_kernel(const float* __restrict__ preds,
                                 float* __restrict__ p0,
                                 float* __restrict__ mask, int n) {
  int t = blockIdx.x * blockDim.x + threadIdx.x;
  if (t < n) { p0[t] = preds[t]; mask[t] = 0.0f; }
}

// ---- scatter seed mask (separate kernel: no race with init) ---------------
__global__ void diff_seed_kernel(const int* __restrict__ seed_idx,
                                 float* __restrict__ mask) {
  int t = blockIdx.x * blockDim.x + threadIdx.x;
  if (t < NSEEDS) {
    int b = seed_idx[2 * t + 0];
    int n = seed_idx[2 * t + 1];
    mask[(size_t)b * NN + n] = 1.0f;
  }
}

// ---- one diffusion iteration via 3 fused fp32 WMMA GEMMs ------------------
// S[i,a] = sum_b [ pred*P + pred^2*P^2/2 + pred^3*P^3/3 ]  (== -log prod)
// out[i,a] = mask ? 1 : 1 - exp(-S[i,a])
__global__ __launch_bounds__(WAVES * 32) void diff_step_kernel(
    const float* __restrict__ cur,   // [BB, NN] pred at iteration start
    const float* __restrict__ P,     // [NN, NN] row-major, P[b*NN + a]
    const float* __restrict__ mask,  // [BB, NN] 1.0 at seeds
    float* __restrict__ nxt) {       // [BB, NN] pred after this iteration
  __shared__ float red[WAVES][32][8];

  const int lane = threadIdx.x & 31;
  const int wave = threadIdx.x >> 5;
  // SGPR copy of the wave id -> scalar (s_cmp/s_cbranch) K-loop, no EXEC
  // masking around the WMMAs.
  const int wave_u = __builtin_amdgcn_readfirstlane(wave);
  const int col0 = blockIdx.x << 4;        // 16 output columns per block
  const int c    = lane & 15;
  const int rsel = lane >> 4;              // lane half selects K sub-pairs
  const float keep = (c < BB) ? 1.0f : 0.0f;  // zero the 8 padding A-rows
  const int arow = c & (BB - 1);

  v8f acc = {};                            // f32 16x16 accumulator (8 VGPRs)
  const int kbase = wave_u * KWAVE;
  const int kend  = kbase + KWAVE;

#pragma unroll 4
  for (int k = kbase; k < kend; k += 4) {
    // stream-ahead prefetch of the P (B-matrix) rows, WGP-scope policy
    // (locality 3 -> pull into all cache levels) -> global_prefetch_b8
    __builtin_prefetch(&P[(size_t)(k + PF_DIST + rsel) * NN + col0 + c], 0, 3);

    // B fragment (4x16, f32): VGPR0 = rows K0(l0-15)/K1(l16-31), VGPR1 = K2/K3
    float b0 = P[(size_t)(k + rsel)     * NN + col0 + c];
    float b1 = P[(size_t)(k + rsel + 2) * NN + col0 + c];
    // A fragment (16x4, f32): VGPR0 = K0(l0-15)/K2(l16-31), VGPR1 = K1/K3
    float a0 = cur[(size_t)arow * NN + (k + 2 * rsel)]     * keep;
    float a1 = cur[(size_t)arow * NN + (k + 2 * rsel + 1)] * keep;

    v2f A1 = {a0, a1};
    v2f B1 = {b0, b1};
    v2f A2 = {a0 * a0, a1 * a1};
    v2f B2 = {0.5f * b0 * b0, 0.5f * b1 * b1};
    v2f A3 = {a0 * a0 * a0, a1 * a1 * a1};
    v2f B3 = {(1.0f / 3.0f) * b0 * b0 * b0, (1.0f / 3.0f) * b1 * b1 * b1};

    acc = __builtin_amdgcn_wmma_f32_16x16x4_f32(false, A1, false, B1,
                                                (short)0, acc, false, false);
    acc = __builtin_amdgcn_wmma_f32_16x16x4_f32(false, A2, false, B2,
                                                (short)0, acc, false, false);
    acc = __builtin_amdgcn_wmma_f32_16x16x4_f32(false, A3, false, B3,
                                                (short)0, acc, false, false);
  }

  // split-K reduction across the 16 waves through LDS
#pragma unroll
  for (int j = 0; j < 8; ++j) red[wave][lane][j] = acc[j];
  __syncthreads();

  // C/D layout: VGPR j, lanes 0-15 -> (M=j, N=lane); rows 0..7 are the real
  // batch rows, so lanes 0..15 of wave 0 own every real output of this tile.
  if (wave == 0 && lane < 16) {
#pragma unroll
    for (int j = 0; j < BB; ++j) {
      float s = 0.0f;
#pragma unroll
      for (int w = 0; w < WAVES; ++w) s += red[w][lane][j];
      float val = 1.0f - __expf(-s);
      size_t o = (size_t)j * NN + col0 + lane;
      nxt[o] = (mask[o] > 0.5f) ? 1.0f : val;
    }
  }
}

extern "C" void kernel_launch(void* const* d_in, const int* in_sizes, int n_in,
                              void* d_out, int out_size, void* d_ws,
                              size_t ws_size, hipStream_t stream) {
  const float* preds    = (const float*)d_in[0];
  const float* P        = (const float*)d_in[1];
  const int*   seed_idx = (const int*)d_in[2];
  float*       out      = (float*)d_out;

  // workspace: [mask BB*NN][p0 BB*NN][p1 BB*NN] = 384 KB
  float* mask = (float*)d_ws;
  float* p0   = mask + (size_t)BB * NN;
  float* p1   = p0 + (size_t)BB * NN;

  const int n = BB * NN;
  diff_init_kernel<<<(n + 255) / 256, 256, 0, stream>>>(preds, p0, mask, n);
  diff_seed_kernel<<<1, 128, 0, stream>>>(seed_idx, mask);

  dim3 grid(NN / 16), block(WAVES * 32);
  diff_step_kernel<<<grid, block, 0, stream>>>(p0, P, mask, p1);
  diff_step_kernel<<<grid, block, 0, stream>>>(p1, P, mask, p0);
  diff_step_kernel<<<grid, block, 0, stream>>>(p0, P, mask, p1);
  diff_step_kernel<<<grid, block, 0, stream>>>(p1, P, mask, out);
}